// singleHeadedLogitDecoder_32564442038589
// MI455X (gfx1250) — compile-verified
//
#include <hip/hip_runtime.h>
#include <hip/hip_bf16.h>
#include <math.h>

// ---------------------------------------------------------------------------
// singleHeadedLogitDecoder for MI455X (gfx1250, wave32)
//
// Math rewrite:  compat[k,i] = x[k,i,:] . q2[k,:]   where
//                q2[k,:] = (Wq @ ctx[k,:])^T @ Wk   (all fp32, exact)
// node_values is dead code in the reference and is skipped.
//
// K1: two fused 128x512x512 fp32 GEMMs via V_WMMA_F32_16X16X4_F32.
// K2: HBM-roofline stream of node_embeddings (512 MB) with NT loads,
//     split across 1024 blocks for full-chip WGP coverage.
// K3: per-row softmax (1 MB traffic).
// Fallback fused K2+K3 if d_ws is too small for the 1 MB logit buffer.
// ---------------------------------------------------------------------------

typedef __attribute__((ext_vector_type(2))) float v2f;
typedef __attribute__((ext_vector_type(4))) float v4f;
typedef __attribute__((ext_vector_type(8))) float v8f;

#define K_ROWS 128
#define N_NODES 2048
#define DIM 512
#define INV_SQRT_D 0.044194173824159216f  // 1/sqrt(512)

// ---------------------------------------------------------------------------
// Kernel 1: q2 = (ctx @ Wq^T) @ Wk     [128,512]
// grid = 8 blocks, block = 1024 threads (32 waves).
// Block b owns M-tile rows [16b,16b+16). Wave w owns N-tile cols [16w,16w+16).
// WMMA f32 16x16x4 layouts (ISA 7.12.2):
//   A 16x4 : lane l (m=l&15, h=l>>4): vgpr0=A[m][2h], vgpr1=A[m][2h+1]
//   B 4x16 : lane l (n=l&15, h=l>>4): vgpr0=B[2h][n], vgpr1=B[2h+1][n]
//   C 16x16: lanes 0-15 hold M=r (vgpr r), lanes 16-31 hold M=8+r.
// ---------------------------------------------------------------------------
__global__ void __launch_bounds__(1024)
qproj_wmma_kernel(const float* __restrict__ ctx,   // [128,512]
                  const float* __restrict__ Wq,    // [512,512]
                  const float* __restrict__ Wk,    // [512,512]
                  float* __restrict__ q2)          // [128,512] (d_ws)
{
    __shared__ __align__(16) float cq_lds[16 * DIM];   // 32 KB

    const int ktile = blockIdx.x;             // 0..7
    const int wave  = threadIdx.x >> 5;       // 0..31 -> N-tile
    const int lane  = threadIdx.x & 31;
    const int m     = lane & 15;
    const int half  = lane >> 4;              // 0 or 1
    const int kk    = half * 2;               // K sub-offset within 4-chunk
    const int n0    = wave * 16;
    const int nn    = m;                      // column within N-tile

    // ---- GEMM1: cq = ctx @ Wq^T  (B[kk][nn] = Wq[n0+nn][k0+kk]) ----
    v8f acc = {};
    const float* arow = ctx + (ktile * 16 + m) * DIM;
    const float* bq   = Wq + (n0 + nn) * DIM;
    for (int k0 = 0; k0 < DIM; k0 += 4) {
        v2f a, b;
        a.x = arow[k0 + kk];
        a.y = arow[k0 + kk + 1];
        b.x = bq[k0 + kk];
        b.y = bq[k0 + kk + 1];
        acc = __builtin_amdgcn_wmma_f32_16x16x4_f32(
            false, a, false, b, (short)0, acc, false, false);
    }
#pragma unroll
    for (int r = 0; r < 8; ++r) {
        cq_lds[(half * 8 + r) * DIM + n0 + nn] = acc[r];
    }
    __syncthreads();

    // ---- GEMM2: q2 = cq @ Wk  (B[kk][nn] = Wk[k0+kk][n0+nn], coalesced) ----
    v8f acc2 = {};
    const float* alds = cq_lds + m * DIM;
    for (int k0 = 0; k0 < DIM; k0 += 4) {
        v2f a, b;
        a.x = alds[k0 + kk];
        a.y = alds[k0 + kk + 1];
        b.x = Wk[(k0 + kk) * DIM + n0 + nn];
        b.y = Wk[(k0 + kk + 1) * DIM + n0 + nn];
        acc2 = __builtin_amdgcn_wmma_f32_16x16x4_f32(
            false, a, false, b, (short)0, acc2, false, false);
    }
#pragma unroll
    for (int r = 0; r < 8; ++r) {
        q2[(ktile * 16 + half * 8 + r) * DIM + n0 + nn] = acc2[r];
    }
}

// ---------------------------------------------------------------------------
// Per-wave dot product of one 512-dim node row against LDS-resident q2,
// with the reference's clamp/tanh/scale/mask epilogue. NT loads: the
// 512 MB stream is touched once and exceeds the 192 MB L2.
// ---------------------------------------------------------------------------
__device__ __forceinline__ float node_logit(const float* __restrict__ x,
                                            const float* __restrict__ s_q,
                                            int lane) {
    v4f s4 = {};
#pragma unroll
    for (int d0 = 0; d0 < DIM; d0 += 128) {
        const v4f xv = __builtin_nontemporal_load((const v4f*)(x + d0 + lane * 4));
        const v4f qv = *(const v4f*)(s_q + d0 + lane * 4);
        s4.x = fmaf(xv.x, qv.x, s4.x);
        s4.y = fmaf(xv.y, qv.y, s4.y);
        s4.z = fmaf(xv.z, qv.z, s4.z);
        s4.w = fmaf(xv.w, qv.w, s4.w);
    }
    float s = (s4.x + s4.y) + (s4.z + s4.w);
#pragma unroll
    for (int off = 16; off > 0; off >>= 1)
        s += __shfl_xor(s, off, 32);
    return s;
}

__device__ __forceinline__ float logit_epilogue(float s, unsigned char msk) {
    float c = fminf(fmaxf(s, -10.0f), 10.0f);
    c = tanhf(c) * INV_SQRT_D;
    return msk ? -INFINITY : c;
}

// ---------------------------------------------------------------------------
// Kernel 2 (split path): logits only. grid = (8 chunks, 128 k) = 1024 blocks,
// block = 256 threads (8 waves). Each block streams 256 node rows (512 KB).
// ---------------------------------------------------------------------------
__global__ void __launch_bounds__(256)
logits_kernel(const float* __restrict__ node,          // [128,2048,512]
              const unsigned char* __restrict__ mask,  // [128,2048] bool
              const float* __restrict__ q2,            // [128,512]
              float* __restrict__ lbuf)                // [128,2048]
{
    __shared__ __align__(16) float s_q[DIM];
    const int k     = blockIdx.y;
    const int base  = blockIdx.x * (N_NODES / 8);      // 256-node chunk
    const int tid   = threadIdx.x;
    const int wave  = tid >> 5;                        // 0..7
    const int lane  = tid & 31;

    s_q[tid]       = q2[k * DIM + tid];
    s_q[tid + 256] = q2[k * DIM + tid + 256];
    __syncthreads();

    for (int i = base + wave; i < base + (N_NODES / 8); i += 8) {
        const float* x = node + ((size_t)k * N_NODES + i) * DIM;
        const float s = node_logit(x, s_q, lane);
        if (lane == 0)
            lbuf[k * N_NODES + i] = logit_epilogue(s, mask[k * N_NODES + i]);
    }
}

// ---------------------------------------------------------------------------
// Kernel 3 (split path): row softmax over 2048 logits.
// grid = 128, block = 512 (4 elements/thread).
// ---------------------------------------------------------------------------
__global__ void __launch_bounds__(512)
softmax_kernel(const float* __restrict__ lbuf,    // [128,2048]
               float* __restrict__ out)           // [128,2048]
{
    __shared__ float s_v[N_NODES];
    __shared__ float s_red[16];

    const int k    = blockIdx.x;
    const int tid  = threadIdx.x;
    const int wave = tid >> 5;
    const int lane = tid & 31;

    float lmax = -INFINITY;
#pragma unroll
    for (int j = 0; j < 4; ++j) {
        const float v = lbuf[k * N_NODES + tid + j * 512];
        s_v[tid + j * 512] = v;
        lmax = fmaxf(lmax, v);
    }
#pragma unroll
    for (int off = 16; off > 0; off >>= 1)
        lmax = fmaxf(lmax, __shfl_xor(lmax, off, 32));
    if (lane == 0) s_red[wave] = lmax;
    __syncthreads();
    if (tid == 0) {
        float mm = s_red[0];
#pragma unroll
        for (int w = 1; w < 16; ++w) mm = fmaxf(mm, s_red[w]);
        s_red[0] = mm;
    }
    __syncthreads();
    const float gmax = s_red[0];
    __syncthreads();

    float lsum = 0.0f;
#pragma unroll
    for (int j = 0; j < 4; ++j) {
        const int idx = tid + j * 512;
        const float e = __expf(s_v[idx] - gmax);
        s_v[idx] = e;
        lsum += e;
    }
#pragma unroll
    for (int off = 16; off > 0; off >>= 1)
        lsum += __shfl_xor(lsum, off, 32);
    if (lane == 0) s_red[wave] = lsum;
    __syncthreads();
    if (tid == 0) {
        float ss = 0.0f;
#pragma unroll
        for (int w = 0; w < 16; ++w) ss += s_red[w];
        s_red[0] = ss;
    }
    __syncthreads();
    const float inv = 1.0f / s_red[0];
#pragma unroll
    for (int j = 0; j < 4; ++j) {
        const int idx = tid + j * 512;
        out[k * N_NODES + idx] = s_v[idx] * inv;
    }
}

// ---------------------------------------------------------------------------
// Fallback fused kernel (used only if d_ws can't hold the logit buffer):
// grid = 128 (one block per k), block = 512 threads (16 waves).
// ---------------------------------------------------------------------------
__global__ void __launch_bounds__(512)
attn_softmax_fused_kernel(const float* __restrict__ node,
                          const unsigned char* __restrict__ mask,
                          const float* __restrict__ q2,
                          float* __restrict__ out)
{
    __shared__ __align__(16) float s_q[DIM];
    __shared__ float s_logit[N_NODES];
    __shared__ float s_red[16];

    const int k    = blockIdx.x;
    const int tid  = threadIdx.x;
    const int wave = tid >> 5;
    const int lane = tid & 31;

    s_q[tid] = q2[k * DIM + tid];
    __syncthreads();

    for (int i = wave; i < N_NODES; i += 16) {
        const float* x = node + ((size_t)k * N_NODES + i) * DIM;
        const float s = node_logit(x, s_q, lane);
        if (lane == 0)
            s_logit[i] = logit_epilogue(s, mask[k * N_NODES + i]);
    }
    __syncthreads();

    float lmax = -INFINITY;
#pragma unroll
    for (int j = 0; j < 4; ++j)
        lmax = fmaxf(lmax, s_logit[tid + j * 512]);
#pragma unroll
    for (int off = 16; off > 0; off >>= 1)
        lmax = fmaxf(lmax, __shfl_xor(lmax, off, 32));
    if (lane == 0) s_red[wave] = lmax;
    __syncthreads();
    if (tid == 0) {
        float mm = s_red[0];
#pragma unroll
        for (int w = 1; w < 16; ++w) mm = fmaxf(mm, s_red[w]);
        s_red[0] = mm;
    }
    __syncthreads();
    const float gmax = s_red[0];
    __syncthreads();

    float lsum = 0.0f;
#pragma unroll
    for (int j = 0; j < 4; ++j) {
        const int idx = tid + j * 512;
        const float e = __expf(s_logit[idx] - gmax);
        s_logit[idx] = e;
        lsum += e;
    }
#pragma unroll
    for (int off = 16; off > 0; off >>= 1)
        lsum += __shfl_xor(lsum, off, 32);
    if (lane == 0) s_red[wave] = lsum;
    __syncthreads();
    if (tid == 0) {
        float ss = 0.0f;
#pragma unroll
        for (int w = 0; w < 16; ++w) ss += s_red[w];
        s_red[0] = ss;
    }
    __syncthreads();
    const float inv = 1.0f / s_red[0];
#pragma unroll
    for (int j = 0; j < 4; ++j) {
        const int idx = tid + j * 512;
        out[k * N_NODES + idx] = s_logit[idx] * inv;
    }
}

// ---------------------------------------------------------------------------
// Launch. Inputs: 0=context_embedding, 1=node_embeddings, 2=mask (bool/u8),
//                 3=Wq, 4=Wk, 5=Wv (unused). Output: [128,2048] f32.
// d_ws layout: [0,256KB) q2 ; [256KB, 256KB+1MB) logits (split path only).
// The ws_size branch is a compile-time-style constant per run: deterministic.
// ---------------------------------------------------------------------------
extern "C" void kernel_launch(void* const* d_in, const int* in_sizes, int n_in,
                              void* d_out, int out_size, void* d_ws, size_t ws_size,
                              hipStream_t stream) {
    (void)in_sizes; (void)n_in; (void)out_size;
    const float* ctx  = (const float*)d_in[0];
    const float* node = (const float*)d_in[1];
    const unsigned char* mask = (const unsigned char*)d_in[2];
    const float* Wq   = (const float*)d_in[3];
    const float* Wk   = (const float*)d_in[4];
    float* q2  = (float*)d_ws;
    float* out = (float*)d_out;

    const size_t q2_bytes   = (size_t)K_ROWS * DIM * sizeof(float);      // 256 KB
    const size_t lbuf_bytes = (size_t)K_ROWS * N_NODES * sizeof(float);  // 1 MB

    qproj_wmma_kernel<<<dim3(K_ROWS / 16), dim3(1024), 0, stream>>>(ctx, Wq, Wk, q2);

    if (ws_size >= q2_bytes + lbuf_bytes) {
        float* lbuf = (float*)((char*)d_ws + q2_bytes);
        logits_kernel<<<dim3(8, K_ROWS), dim3(256), 0, stream>>>(node, mask, q2, lbuf);
        softmax_kernel<<<dim3(K_ROWS), dim3(512), 0, stream>>>(lbuf, out);
    } else {
        attn_softmax_fused_kernel<<<dim3(K_ROWS), dim3(512), 0, stream>>>(node, mask, q2, out);
    }
}